// Multihead_Attention_16011638080099
// MI455X (gfx1250) — compile-verified
//
#include <hip/hip_runtime.h>
#include <math.h>

typedef __attribute__((ext_vector_type(2))) float v2f;
typedef __attribute__((ext_vector_type(8))) float v8f;

#define NBATCH 4
#define CIN 512
#define HH 48
#define WW 48
#define LSPAT (HH*WW)      // 2304
#define CH3 192            // 3 * 8 heads * 8 dim
#define HCH 64             // 8 heads * 8 dim
#define NHEAD 8
#define DHEAD 8
#define ATTN_SCALE 0.35355339059327379f  // 1/sqrt(8)

// D = A(16x4 f32) * B(4x16 f32) + C, full fp32 on the matrix pipe.
static __device__ __forceinline__ v8f wmma4(v2f a, v2f b, v8f c) {
  return __builtin_amdgcn_wmma_f32_16x16x4_f32(false, a, false, b, (short)0, c,
                                               false, false);
}

// ---------------------------------------------------------------------------
// K1: qkv 1x1 conv as GEMM: qkv[n][192][L] = W[192x512] * x[n][512][L] + b
// One wave per 16x16 output tile; 128 WMMA steps over K=512.
// ---------------------------------------------------------------------------
__global__ __launch_bounds__(256) void qkv_gemm_kernel(
    const float* __restrict__ x, const float* __restrict__ w,
    const float* __restrict__ bias, float* __restrict__ qkv) {
  const int lane = threadIdx.x & 31;
  const int half = lane >> 4;      // 0: lanes 0-15, 1: lanes 16-31
  const int l16  = lane & 15;
  const int wave = blockIdx.x * (blockDim.x >> 5) + (threadIdx.x >> 5);

  const int ntn = LSPAT / 16;      // 144 col tiles
  const int ntm = CH3 / 16;        // 12 row tiles
  int t = wave;
  const int n  = t / (ntm * ntn);  t -= n * ntm * ntn;
  const int mt = t / ntn;
  const int ct = t % ntn;

  const int mrow = mt * 16 + l16;  // A: M = lane&15
  const int col  = ct * 16 + l16;  // B: N = lane&15
  const float* xb = x + (size_t)n * CIN * LSPAT;

  v8f acc = {};
  for (int kk = 0; kk < CIN; kk += 4) {
    const int ka = kk + half * 2;  // K = 2*(lane>=16) + j
    v2f a; a[0] = w[mrow * CIN + ka];
           a[1] = w[mrow * CIN + ka + 1];
    v2f b; b[0] = xb[(size_t)ka * LSPAT + col];
           b[1] = xb[(size_t)(ka + 1) * LSPAT + col];
    acc = wmma4(a, b, acc);
  }

  float* out = qkv + (size_t)n * CH3 * LSPAT;
#pragma unroll
  for (int r = 0; r < 8; ++r) {    // D: M = r + 8*(lane>=16), N = lane&15
    const int row = mt * 16 + r + half * 8;
    out[(size_t)row * LSPAT + col] = acc[r] + bias[row];
  }
}

// ---------------------------------------------------------------------------
// K2: fused attention per (n,h). Each wave owns 16 q rows; K/V stay in L2.
// Three streams over the 144 k-tiles:
//   (a) scores -> elementwise running max (no per-tile shuffles)
//   (b) scores -> sum of exp((s-m)*scale)
//   (c) scores -> normalized P -> LDS re-fragment -> WMMA P*V^T
// Row-wide butterflies happen exactly twice (after a and after b).
// The 2304x2304 attention matrix never reaches global memory.
// ---------------------------------------------------------------------------
__global__ __launch_bounds__(128) void attn_kernel(
    const float* __restrict__ qkv, float* __restrict__ obuf) {
  __shared__ float plds[4][16][17];
  const int lane = threadIdx.x & 31;
  const int half = lane >> 4;
  const int l16  = lane & 15;
  const int wv   = threadIdx.x >> 5;

  const int qtiles = LSPAT / 64;              // 36 blocks of 64 q each
  const int nh = blockIdx.x / qtiles;
  const int qt = blockIdx.x % qtiles;
  const int n = nh >> 3, h = nh & 7;
  const int qbase = qt * 64 + wv * 16;

  const float* base = qkv + (size_t)n * CH3 * LSPAT + (size_t)h * 24 * LSPAT;
  const float* qm = base;                          // q[d][col]
  const float* km = base + (size_t)DHEAD * LSPAT;  // k[d][col]
  const float* vm = base + (size_t)2 * DHEAD * LSPAT;

  // Q A-fragments (fixed for the whole k loop): A[m=q, K=d]
  v2f aq[2];
#pragma unroll
  for (int db = 0; db < 2; ++db) {
    const int d0 = db * 4 + half * 2;
    aq[db][0] = qm[(size_t)d0 * LSPAT + qbase + l16];
    aq[db][1] = qm[(size_t)(d0 + 1) * LSPAT + qbase + l16];
  }

  // ---- Stream (a): elementwise running max of raw scores -------------------
  float m8[8];
#pragma unroll
  for (int r = 0; r < 8; ++r) m8[r] = -3.0e38f;

  for (int kt = 0; kt < LSPAT / 16; ++kt) {
    const int kb = kt * 16;
    v8f s = {};
#pragma unroll
    for (int db = 0; db < 2; ++db) {
      const int d0 = db * 4 + half * 2;
      v2f bk; bk[0] = km[(size_t)d0 * LSPAT + kb + l16];
              bk[1] = km[(size_t)(d0 + 1) * LSPAT + kb + l16];
      s = wmma4(aq[db], bk, s);
    }
#pragma unroll
    for (int r = 0; r < 8; ++r) m8[r] = fmaxf(m8[r], s[r]);
  }
  // One butterfly per row (row = r + 8*half spans its 16-lane half).
#pragma unroll
  for (int r = 0; r < 8; ++r) {
#pragma unroll
    for (int m = 1; m < 16; m <<= 1) m8[r] = fmaxf(m8[r], __shfl_xor(m8[r], m, 32));
  }

  // ---- Stream (b): row sums of exp((s-m)*scale) ----------------------------
  float s8[8];
#pragma unroll
  for (int r = 0; r < 8; ++r) s8[r] = 0.0f;

  for (int kt = 0; kt < LSPAT / 16; ++kt) {
    const int kb = kt * 16;
    v8f s = {};
#pragma unroll
    for (int db = 0; db < 2; ++db) {
      const int d0 = db * 4 + half * 2;
      v2f bk; bk[0] = km[(size_t)d0 * LSPAT + kb + l16];
              bk[1] = km[(size_t)(d0 + 1) * LSPAT + kb + l16];
      s = wmma4(aq[db], bk, s);
    }
#pragma unroll
    for (int r = 0; r < 8; ++r) s8[r] += __expf((s[r] - m8[r]) * ATTN_SCALE);
  }
  float inv[8];
#pragma unroll
  for (int r = 0; r < 8; ++r) {
#pragma unroll
    for (int m = 1; m < 16; m <<= 1) s8[r] += __shfl_xor(s8[r], m, 32);
    inv[r] = 1.0f / s8[r];
  }

  // ---- Stream (c): normalize P and accumulate out = P * V^T ----------------
  v8f acc = {};
  for (int kt = 0; kt < LSPAT / 16; ++kt) {
    const int kb = kt * 16;
    v8f s = {};
#pragma unroll
    for (int db = 0; db < 2; ++db) {
      const int d0 = db * 4 + half * 2;
      v2f bk; bk[0] = km[(size_t)d0 * LSPAT + kb + l16];
              bk[1] = km[(size_t)(d0 + 1) * LSPAT + kb + l16];
      s = wmma4(aq[db], bk, s);
    }
#pragma unroll
    for (int r = 0; r < 8; ++r)       // D-layout -> LDS (per-wave region)
      plds[wv][r + half * 8][l16] =
          __expf((s[r] - m8[r]) * ATTN_SCALE) * inv[r];
    __syncthreads();
#pragma unroll
    for (int c = 0; c < 4; ++c) {     // K=16 split into 4 WMMA steps
      const int k0 = c * 4 + half * 2;
      v2f ap; ap[0] = plds[wv][l16][k0];       // A[m=q, K=k]
              ap[1] = plds[wv][l16][k0 + 1];
      v2f bv;                                  // B[K=k, N=d] (d<8 valid)
      if (l16 < 8) {
        bv[0] = vm[(size_t)l16 * LSPAT + kb + k0];
        bv[1] = vm[(size_t)l16 * LSPAT + kb + k0 + 1];
      } else { bv[0] = 0.0f; bv[1] = 0.0f; }
      acc = wmma4(ap, bv, acc);
    }
    __syncthreads();
  }

  if (l16 < 8) {                      // D: M=q row, N=d col
    float* ob = obuf + (size_t)n * HCH * LSPAT + (size_t)(h * DHEAD + l16) * LSPAT;
#pragma unroll
    for (int r = 0; r < 8; ++r) ob[qbase + r + half * 8] = acc[r];
  }
}

// ---------------------------------------------------------------------------
// K3: pos 3x3 SAME conv on v (64->64 ch), accumulated into obuf. 0.34 GFLOP.
// ---------------------------------------------------------------------------
__global__ __launch_bounds__(256) void pos_conv_kernel(
    const float* __restrict__ qkv, const float* __restrict__ pw,
    const float* __restrict__ pb, float* __restrict__ obuf) {
  int idx = blockIdx.x * 256 + threadIdx.x;   // n * 64 * 2304 threads
  const int xp = idx % WW; int t = idx / WW;
  const int yp = t % HH;  t /= HH;
  const int oc = t % HCH; const int n = t / HCH;

  const float* vbase = qkv + (size_t)n * CH3 * LSPAT;
  float acc = pb[oc];
  for (int ic = 0; ic < HCH; ++ic) {
    const int ch = (ic >> 3) * 24 + 16 + (ic & 7);  // v channel in qkv layout
    const float* vin  = vbase + (size_t)ch * LSPAT;
    const float* wrow = pw + ((size_t)oc * HCH + ic) * 9;
#pragma unroll
    for (int ky = 0; ky < 3; ++ky) {
      const int iy = yp + ky - 1;
      if (iy < 0 || iy >= HH) continue;
#pragma unroll
      for (int kx = 0; kx < 3; ++kx) {
        const int ix = xp + kx - 1;
        if (ix < 0 || ix >= WW) continue;
        acc = fmaf(vin[iy * WW + ix], wrow[ky * 3 + kx], acc);
      }
    }
  }
  obuf[(size_t)n * HCH * LSPAT + (size_t)oc * LSPAT + yp * WW + xp] += acc;
}

// ---------------------------------------------------------------------------
// K4: res 1x1 conv as GEMM: out[n][512][L] = Wr[512x64] * obuf[n][64][L] + b
// ---------------------------------------------------------------------------
__global__ __launch_bounds__(256) void res_gemm_kernel(
    const float* __restrict__ obuf, const float* __restrict__ w,
    const float* __restrict__ bias, float* __restrict__ out) {
  const int lane = threadIdx.x & 31;
  const int half = lane >> 4;
  const int l16  = lane & 15;
  const int wave = blockIdx.x * (blockDim.x >> 5) + (threadIdx.x >> 5);

  const int ntn = LSPAT / 16;   // 144
  const int ntm = CIN / 16;     // 32
  int t = wave;
  const int n  = t / (ntm * ntn); t -= n * ntm * ntn;
  const int mt = t / ntn;
  const int ct = t % ntn;

  const int mrow = mt * 16 + l16;
  const int col  = ct * 16 + l16;
  const float* ib = obuf + (size_t)n * HCH * LSPAT;

  v8f acc = {};
#pragma unroll
  for (int kk = 0; kk < HCH; kk += 4) {
    const int ka = kk + half * 2;
    v2f a; a[0] = w[mrow * HCH + ka];
           a[1] = w[mrow * HCH + ka + 1];
    v2f b; b[0] = ib[(size_t)ka * LSPAT + col];
           b[1] = ib[(size_t)(ka + 1) * LSPAT + col];
    acc = wmma4(a, b, acc);
  }

  float* ob = out + (size_t)n * CIN * LSPAT;
#pragma unroll
  for (int r = 0; r < 8; ++r) {
    const int row = mt * 16 + r + half * 8;
    ob[(size_t)row * LSPAT + col] = acc[r] + bias[row];
  }
}

extern "C" void kernel_launch(void* const* d_in, const int* in_sizes, int n_in,
                              void* d_out, int out_size, void* d_ws, size_t ws_size,
                              hipStream_t stream) {
  const float* x     = (const float*)d_in[0];
  const float* qkv_w = (const float*)d_in[1];
  const float* qkv_b = (const float*)d_in[2];
  const float* pos_w = (const float*)d_in[3];
  const float* pos_b = (const float*)d_in[4];
  const float* res_w = (const float*)d_in[5];
  const float* res_b = (const float*)d_in[6];
  float* out = (float*)d_out;

  float* qkv  = (float*)d_ws;                       // 4*192*2304 f32 = 7.08 MB
  float* obuf = qkv + (size_t)NBATCH * CH3 * LSPAT; // 4*64*2304  f32 = 2.36 MB

  const int waves1 = NBATCH * (CH3 / 16) * (LSPAT / 16);  // 6912 waves
  qkv_gemm_kernel<<<waves1 / 8, 256, 0, stream>>>(x, qkv_w, qkv_b, qkv);

  attn_kernel<<<NBATCH * NHEAD * (LSPAT / 64), 128, 0, stream>>>(qkv, obuf);

  pos_conv_kernel<<<(NBATCH * HCH * LSPAT) / 256, 256, 0, stream>>>(
      qkv, pos_w, pos_b, obuf);

  const int waves4 = NBATCH * (CIN / 16) * (LSPAT / 16);  // 18432 waves
  res_gemm_kernel<<<waves4 / 8, 256, 0, stream>>>(obuf, res_w, res_b, out);
}